// EphysAttentionLayer_74457553043516
// MI455X (gfx1250) — compile-verified
//
#include <hip/hip_runtime.h>

typedef __attribute__((ext_vector_type(2))) float v2f;
typedef __attribute__((ext_vector_type(8))) float v8f;

#define DTC 0.001f
#define NN 1024
#define EE 512
#define BATCH 8
#define ROWS 64          // i-rows per block
#define LDS_STRIDE 1028  // floats per padded LDS row (bank-conflict-free b64 A-frag reads)

// ---------- prep 1: batch-invariant exp maps over N*N ----------
__global__ __launch_bounds__(256) void prep_maps(
    const float* __restrict__ lpw, const float* __restrict__ lpostw,
    const float* __restrict__ tau_pre, const float* __restrict__ tau_post,
    float* __restrict__ pre_decay, float* __restrict__ post_decay,
    float* __restrict__ pre_wdt, float* __restrict__ post_wdt) {
  int idx = blockIdx.x * blockDim.x + threadIdx.x;
  if (idx >= NN * NN) return;
  // exp(-DT / exp(tau)) == exp(-DT * exp(-tau))
  pre_decay[idx]  = __expf(-DTC * __expf(-tau_pre[idx]));
  post_decay[idx] = __expf(-DTC * __expf(-tau_post[idx]));
  pre_wdt[idx]    = __expf(lpw[idx]) * DTC;
  post_wdt[idx]   = __expf(lpostw[idx]) * DTC;
}

// ---------- prep 2: v_w -> WMMA B-fragment layout ----------
// Bp[q*EE + e] = { v_w[e*NN + 2q], v_w[e*NN + 2q + 1] },  q in [0, NN/2)
// B-frag for k-step kk, half h: lane reads Bp[(2kk + h)*EE + e] (one b64 load).
__global__ __launch_bounds__(256) void prep_bp(const float* __restrict__ vw,
                                               float2* __restrict__ bp) {
  int idx = blockIdx.x * blockDim.x + threadIdx.x;
  if (idx >= (NN / 2) * EE) return;
  int e = idx / (NN / 2);   // coalesced reads along v_w rows
  int q = idx - e * (NN / 2);
  bp[(size_t)q * EE + e] = make_float2(vw[(size_t)e * NN + 2 * q],
                                       vw[(size_t)e * NN + 2 * q + 1]);
}

// ---------- fused attention update + fp32 WMMA GEMM ----------
__global__ __launch_bounds__(256) void ephys_attn_gemm(
    const int* __restrict__ spikes,
    const float* __restrict__ pre_trace, const float* __restrict__ post_trace,
    const float* __restrict__ attention,
    const float* __restrict__ pre_decay, const float* __restrict__ post_decay,
    const float* __restrict__ pre_wdt, const float* __restrict__ post_wdt,
    const float2* __restrict__ bp, const float* __restrict__ v_b,
    float* __restrict__ out) {
  extern __shared__ float a_lds[];  // ROWS x LDS_STRIDE floats

  const int blk = blockIdx.x;
  const int b  = blk >> 4;           // NN/ROWS == 16 stripes per batch
  const int i0 = (blk & 15) * ROWS;
  const int tid = threadIdx.x;

  // ---- phase 1: updated attention tile (64 x 1024) -> LDS ----
  const size_t base_bnn = ((size_t)b * NN + i0) * NN;
  for (int idx = tid; idx < ROWS * NN; idx += 256) {
    int r = idx >> 10;          // / NN
    int j = idx & (NN - 1);
    size_t g = base_bnn + idx;                  // pre/post trace, attention
    size_t m = (size_t)(i0 + r) * NN + j;       // batch-invariant maps
    float si = (float)spikes[b * NN + i0 + r];
    float sj = (float)spikes[b * NN + j];
    float preT  = pre_trace[g]  * pre_decay[m]  + sj * pre_wdt[m];
    float postT = post_trace[g] * post_decay[m] + si * post_wdt[m];
    float a = attention[g];
    float an = a + (1.0f - a) * preT  * (si != 0.0f ? 1.0f : 0.0f)
                 -         a  * postT * (sj != 0.0f ? 1.0f : 0.0f);
    an = fminf(fmaxf(an, -0.5f), 1.5f);
    a_lds[r * LDS_STRIDE + j] = an;
  }
  __syncthreads();

  // ---- phase 2: 8 waves = 4 row-groups x 2 col-groups, 16 acc tiles/wave ----
  const int lane = tid & 31;
  const int w    = tid >> 5;
  const int wr   = w >> 1;      // 0..3 -> rows [wr*16, wr*16+16)
  const int wc   = w & 1;       // 0..1 -> cols [wc*256, wc*256+256)
  const int half = lane >> 4;   // K sub-pair selector (A: K+2h, B: row pair 2kk+h)
  const int l15  = lane & 15;

  v8f acc[16];
#pragma unroll
  for (int t = 0; t < 16; ++t) { v8f z = {}; acc[t] = z; }

  const float* arow = a_lds + (wr * 16 + l15) * LDS_STRIDE + 2 * half;
  const float2* bprow = bp + (size_t)half * EE + wc * 256 + l15;

  for (int kk = 0; kk < NN / 4; ++kk) {
    v2f afrag = *(const v2f*)(arow + 4 * kk);              // ds_load_b64
    const float2* bbase = bprow + (size_t)(2 * kk) * EE;
    // locality 3 -> WGP-scope prefetch: pulls next k-step's B stream into WGP$
    __builtin_prefetch((const void*)(bbase + 2 * EE), 0, 3);
#pragma unroll
    for (int t = 0; t < 16; ++t) {
      float2 bv = bbase[t * 16];                           // global_load_b64 (L2/L0)
      v2f bfrag; bfrag.x = bv.x; bfrag.y = bv.y;
      acc[t] = __builtin_amdgcn_wmma_f32_16x16x4_f32(
          /*neg_a=*/false, afrag, /*neg_b=*/false, bfrag,
          /*c_mod=*/(short)0, acc[t], /*reuse_a=*/false, /*reuse_b=*/false);
    }
  }

  // ---- epilogue: C/D layout -> out[b,i,e] + bias ----
  const size_t out_base = ((size_t)b * NN + i0 + wr * 16 + half * 8) * EE;
#pragma unroll
  for (int t = 0; t < 16; ++t) {
    int e = wc * 256 + t * 16 + l15;
    float bias = v_b[e];
#pragma unroll
    for (int vg = 0; vg < 8; ++vg)
      out[out_base + (size_t)vg * EE + e] = acc[t][vg] + bias;
  }
}

extern "C" void kernel_launch(void* const* d_in, const int* in_sizes, int n_in,
                              void* d_out, int out_size, void* d_ws, size_t ws_size,
                              hipStream_t stream) {
  const int*   spikes     = (const int*)  d_in[0];
  const float* pre_trace  = (const float*)d_in[1];
  const float* post_trace = (const float*)d_in[2];
  const float* attention  = (const float*)d_in[3];
  const float* lpw        = (const float*)d_in[4];
  const float* lpostw     = (const float*)d_in[5];
  const float* tau_pre    = (const float*)d_in[6];
  const float* tau_post   = (const float*)d_in[7];
  const float* v_w        = (const float*)d_in[8];
  const float* v_b        = (const float*)d_in[9];

  // workspace layout: Bp (2MB) | pre_decay | post_decay | pre_wdt | post_wdt (4MB each)
  char* ws = (char*)d_ws;
  float2* bp      = (float2*)ws;
  float* pre_dec  = (float*)(ws + (size_t)2 * 1024 * 1024);
  float* post_dec = pre_dec + (size_t)NN * NN;
  float* pre_wdt  = post_dec + (size_t)NN * NN;
  float* post_wdt = pre_wdt + (size_t)NN * NN;

  prep_maps<<<(NN * NN + 255) / 256, 256, 0, stream>>>(
      lpw, lpostw, tau_pre, tau_post, pre_dec, post_dec, pre_wdt, post_wdt);
  prep_bp<<<((NN / 2) * EE + 255) / 256, 256, 0, stream>>>(v_w, bp);

  size_t lds_bytes = (size_t)ROWS * LDS_STRIDE * sizeof(float);  // 263,168 B (< 320KB WGP LDS)
  ephys_attn_gemm<<<BATCH * (NN / ROWS), 256, lds_bytes, stream>>>(
      spikes, pre_trace, post_trace, attention,
      pre_dec, post_dec, pre_wdt, post_wdt, bp, v_b, (float*)d_out);
}